// KNO_DIFFUSION_REACTION_19224273617247
// MI455X (gfx1250) — compile-verified
//
#include <hip/hip_runtime.h>

// ---------------------------------------------------------------------------
// KNO diffusion-reaction for MI455X (gfx1250, wave32).
// GEMMs use V_WMMA_F32_16X16X4_F32 (f32 in, f32 acc) - one wave per 16x16 tile.
// Integral transforms / GP readout are fused exp+FMA loops (VALU/trans bound).
// GP solve reduced from 16384-RHS to 1-RHS via symmetry of Kqq.
// ---------------------------------------------------------------------------

typedef float v2f __attribute__((ext_vector_type(2)));
typedef float v8f __attribute__((ext_vector_type(8)));

#define Q_N 1024
#define C_N 64
#define M_N 16384
#define DEPTH_N 4

__device__ __forceinline__ float gelu_tanh(float x) {
  // jax.nn.gelu approximate=True
  float x3 = x * x * x;
  return 0.5f * x * (1.0f + tanhf(0.7978845608028654f * (x + 0.044715f * x3)));
}

// --------------------------- pairwise sq-dist (Q x Q) ----------------------
__global__ void k_sqd(const float* __restrict__ qn, float* __restrict__ sqd) {
  int idx = blockIdx.x * blockDim.x + threadIdx.x;   // Q*Q threads
  int i = idx >> 10, j = idx & 1023;
  float dx = qn[2 * i] - qn[2 * j];
  float dy = qn[2 * i + 1] - qn[2 * j + 1];
  sqd[idx] = dx * dx + dy * dy;
}

// --------------------------- lift: gelu([f_x,qn] @ W.T + b) ----------------
__global__ void k_lift(const float* __restrict__ fx, const float* __restrict__ qn,
                       const float* __restrict__ W, const float* __restrict__ b,
                       float* __restrict__ fq) {
  int idx = blockIdx.x * blockDim.x + threadIdx.x;   // Q*C threads
  int q = idx >> 6, c = idx & 63;
  float v = fx[q] * W[c * 3 + 0] + qn[2 * q] * W[c * 3 + 1] +
            qn[2 * q + 1] * W[c * 3 + 2] + b[c];
  fq[idx] = gelu_tanh(v);
}

// ------------- Y[1024x64] = (gelu?)( X[1024x64] @ W[64x64]^T + b ) ---------
// One wave per 16x16 output tile. 256 tiles -> grid 64 blocks x 128 threads.
__global__ void k_gemm64_wmma(const float* __restrict__ X, const float* __restrict__ W,
                              const float* __restrict__ b, float* __restrict__ Y,
                              int applyGelu) {
  int wave = threadIdx.x >> 5;
  int lane = threadIdx.x & 31;
  int tile = blockIdx.x * 4 + wave;          // wave-uniform
  int tm = tile >> 2, tn = tile & 3;         // 64 M-tiles x 4 N-tiles
  int m0 = tm * 16, n0 = tn * 16;
  int half = lane >> 4;                      // K pair select
  int l = lane & 15;

  float bias = b[n0 + l];                    // C/D: col = n0+l in both halves
  v8f acc;
#pragma unroll
  for (int r = 0; r < 8; ++r) acc[r] = bias;

  const float* xr = X + (m0 + l) * 64;       // A row, M = m0+l
  const float* wr = W + (n0 + l) * 64;       // B col n = n0+l -> W row (Y=X@W^T)
#pragma unroll
  for (int k0 = 0; k0 < 64; k0 += 4) {
    int ka = k0 + 2 * half;                  // lanes 0-15: K=k0,k0+1 ; 16-31: K=k0+2,k0+3
    v2f a, bb;
    a.x = xr[ka];  a.y = xr[ka + 1];
    bb.x = wr[ka]; bb.y = wr[ka + 1];
    acc = __builtin_amdgcn_wmma_f32_16x16x4_f32(
        /*neg_a=*/false, a, /*neg_b=*/false, bb,
        /*c_mod=*/(short)0, acc, /*reuse_a=*/false, /*reuse_b=*/false);
  }

#pragma unroll
  for (int r = 0; r < 8; ++r) {              // VGPR r: M=r (lanes 0-15), M=r+8 (16-31)
    int row = m0 + r + 8 * half;
    float v = acc[r];
    if (applyGelu) v = gelu_tanh(v);
    Y[row * 64 + n0 + l] = v;
  }
}

// ---- out[k,c] = (gelu?)( skip[k,c] + var[c] * sum_q exp(-sqd[k,q]*g[c]) * fq[q,c]*w[q] )
__global__ void k_integ(const float* __restrict__ sqd, const float* __restrict__ fq,
                        const float* __restrict__ w, const float* __restrict__ skip,
                        const float* __restrict__ log_ls, const float* __restrict__ log_var,
                        float* __restrict__ out, int applyGelu) {
  int c = threadIdx.x & 63;
  int k = blockIdx.x * (blockDim.x >> 6) + (threadIdx.x >> 6);
  float g   = 0.5f * __expf(-2.0f * log_ls[c]);
  float var = __expf(log_var[c]);
  const float* srow = sqd + (size_t)k * Q_N;
  float acc = 0.0f;
  for (int q = 0; q < Q_N; ++q) {
    float hq = fq[q * 64 + c] * w[q];
    acc = fmaf(__expf(-srow[q] * g), hq, acc);
  }
  float v = skip[k * 64 + c] + var * acc;
  if (applyGelu) v = gelu_tanh(v);
  out[k * 64 + c] = v;
}

// --------------------------- final 64 -> 1 projection ----------------------
__global__ void k_proj3(const float* __restrict__ X, const float* __restrict__ W3,
                        const float* __restrict__ b3, float* __restrict__ fvec) {
  int q = blockIdx.x * blockDim.x + threadIdx.x;   // 1024 threads
  float acc = b3[0];
#pragma unroll
  for (int c = 0; c < 64; ++c) acc = fmaf(X[q * 64 + c], W3[c], acc);
  fvec[q] = acc;
}

// --------------------------- Kqq = var*exp(-sqd*g) + jitter*I --------------
__global__ void k_kqq(const float* __restrict__ sqd, const float* __restrict__ out_log_ls,
                      const float* __restrict__ out_log_var, float* __restrict__ Kqq) {
  int idx = blockIdx.x * blockDim.x + threadIdx.x;  // Q*Q threads
  int i = idx >> 10, j = idx & 1023;
  float g   = 0.5f * __expf(-2.0f * out_log_ls[0]);
  float var = __expf(out_log_var[0]);
  float v = var * __expf(-sqd[idx] * g);
  if (i == j) v += 1e-5f;
  Kqq[idx] = v;
}

// ----------- single-block Cholesky + L y = f, L^T x = y (n = 1024) ---------
__global__ void k_chol_solve(float* __restrict__ A, const float* __restrict__ fvec,
                             float* __restrict__ alpha) {
  const int n = Q_N;
  int tid = threadIdx.x;
  __shared__ float piv;
  alpha[tid] = fvec[tid];
  __syncthreads();
  // Cholesky (lower triangle of A)
  for (int k = 0; k < n; ++k) {
    if (tid == k) { piv = sqrtf(A[k * n + k]); A[k * n + k] = piv; }
    __syncthreads();
    float lkk = piv;
    if (tid > k) A[tid * n + k] /= lkk;
    __syncthreads();
    if (tid > k) {
      float aik = A[tid * n + k];
      for (int j = k + 1; j <= tid; ++j) A[tid * n + j] -= aik * A[j * n + k];
    }
    __syncthreads();
  }
  // forward: L y = f
  for (int k = 0; k < n; ++k) {
    if (tid == k) alpha[k] /= A[k * n + k];
    __syncthreads();
    float yk = alpha[k];
    if (tid > k) alpha[tid] -= A[tid * n + k] * yk;
    __syncthreads();
  }
  // backward: L^T x = y
  for (int k = n - 1; k >= 0; --k) {
    if (tid == k) alpha[k] /= A[k * n + k];
    __syncthreads();
    float xk = alpha[k];
    if (tid < k) alpha[tid] -= A[k * n + tid] * xk;
    __syncthreads();
  }
}

// --------------- f_y[m] = var * sum_q exp(-||qn_q - y_m||^2 g) a[q] --------
__global__ void k_gp_out(const float* __restrict__ qn, const float* __restrict__ yg,
                         const float* __restrict__ alpha, const float* __restrict__ out_log_ls,
                         const float* __restrict__ out_log_var, float* __restrict__ fy) {
  __shared__ float sx[Q_N], sy[Q_N], sa[Q_N];
  for (int i = threadIdx.x; i < Q_N; i += blockDim.x) {
    sx[i] = qn[2 * i]; sy[i] = qn[2 * i + 1]; sa[i] = alpha[i];
  }
  __syncthreads();
  int m = blockIdx.x * blockDim.x + threadIdx.x;   // 16384 threads
  float gx = yg[2 * m], gy = yg[2 * m + 1];
  float g   = 0.5f * __expf(-2.0f * out_log_ls[0]);
  float var = __expf(out_log_var[0]);
  float acc = 0.0f;
  for (int q = 0; q < Q_N; ++q) {
    float dx = sx[q] - gx, dy = sy[q] - gy;
    acc = fmaf(__expf(-(dx * dx + dy * dy) * g), sa[q], acc);
  }
  fy[m] = var * acc;
}

// ---------------------------------------------------------------------------
extern "C" void kernel_launch(void* const* d_in, const int* in_sizes, int n_in,
                              void* d_out, int out_size, void* d_ws, size_t ws_size,
                              hipStream_t stream) {
  const float* f_x       = (const float*)d_in[0];
  const float* y_grid    = (const float*)d_in[1];
  const float* q_nodes   = (const float*)d_in[2];
  const float* q_weights = (const float*)d_in[3];
  const float* lift_W    = (const float*)d_in[4];
  const float* lift_b    = (const float*)d_in[5];
  const float* conv_W    = (const float*)d_in[6];
  const float* conv_b    = (const float*)d_in[7];
  const float* int_log_ls  = (const float*)d_in[8];
  const float* int_log_var = (const float*)d_in[9];
  const float* proj_W1   = (const float*)d_in[10];
  const float* proj_b1   = (const float*)d_in[11];
  const float* proj_W2   = (const float*)d_in[12];
  const float* proj_b2   = (const float*)d_in[13];
  const float* proj_W3   = (const float*)d_in[14];
  const float* proj_b3   = (const float*)d_in[15];
  const float* out_log_ls  = (const float*)d_in[16];
  const float* out_log_var = (const float*)d_in[17];
  float* fy = (float*)d_out;

  // workspace carve-up (floats)
  float* ws   = (float*)d_ws;
  float* sqd  = ws;                         // 1024*1024
  float* Kqq  = sqd + Q_N * Q_N;            // 1024*1024
  float* fqA  = Kqq + Q_N * Q_N;            // 1024*64
  float* fqB  = fqA + Q_N * C_N;            // 1024*64
  float* skip = fqB + Q_N * C_N;            // 1024*64
  float* fvec = skip + Q_N * C_N;           // 1024
  float* alpha = fvec + Q_N;                // 1024

  // 1) pairwise sq-dist among quadrature nodes (reused by all layers + Kqq)
  k_sqd<<<(Q_N * Q_N) / 256, 256, 0, stream>>>(q_nodes, sqd);

  // 2) lift
  k_lift<<<(Q_N * C_N) / 256, 256, 0, stream>>>(f_x, q_nodes, lift_W, lift_b, fqA);

  // 3) DEPTH kernel-integral layers
  float* cur = fqA; float* nxt = fqB;
  for (int i = 0; i < DEPTH_N; ++i) {
    k_gemm64_wmma<<<64, 128, 0, stream>>>(cur, conv_W + i * C_N * C_N,
                                          conv_b + i * C_N, skip, /*gelu=*/0);
    k_integ<<<Q_N / 4, 256, 0, stream>>>(sqd, cur, q_weights, skip,
                                         int_log_ls + i * C_N, int_log_var + i * C_N,
                                         nxt, /*gelu=*/(i < DEPTH_N - 1) ? 1 : 0);
    float* t = cur; cur = nxt; nxt = t;
  }

  // 4) projection MLP
  k_gemm64_wmma<<<64, 128, 0, stream>>>(cur, proj_W1, proj_b1, nxt, /*gelu=*/1);
  k_gemm64_wmma<<<64, 128, 0, stream>>>(nxt, proj_W2, proj_b2, skip, /*gelu=*/1);
  k_proj3<<<Q_N / 256, 256, 0, stream>>>(skip, proj_W3, proj_b3, fvec);

  // 5) GP readout: alpha = Kqq^{-1} f, f_y = Kqy^T alpha  (symmetry of Kqq)
  k_kqq<<<(Q_N * Q_N) / 256, 256, 0, stream>>>(sqd, out_log_ls, out_log_var, Kqq);
  k_chol_solve<<<1, Q_N, 0, stream>>>(Kqq, fvec, alpha);
  k_gp_out<<<M_N / 256, 256, 0, stream>>>(q_nodes, y_grid, alpha,
                                          out_log_ls, out_log_var, fy);
}